// FIURIModule_59751585022659
// MI455X (gfx1250) — compile-verified
//
#include <hip/hip_runtime.h>

#define BATCH 16            // B is structurally 16 in the reference
#define CLAMP_MIN_F (-10.0f)
#define CLAMP_MAX_F (10.0f)
#define EPS_F (1e-6f)

typedef float f4 __attribute__((ext_vector_type(4)));

// ---------------------------------------------------------------------------
// FAST PATH edge scatter (placed FIRST in the module so the disasm snippet
// shows it: verify global_atomic_add_f32, 4x global_load_b128 per operand).
// Transposed operands: per edge one 64B line of o_preT, one of Et, and 16
// contiguous f32 atomics into one 64B line of gjT -> 3 L2 lines/edge vs 48
// in the naive (B,N) layout.
// ---------------------------------------------------------------------------
__global__ __launch_bounds__(256) void edge_scatter_t(
    const float* __restrict__ o_preT,  // (N,16)
    const float* __restrict__ Et,      // (N,16)
    const float* __restrict__ w,
    const int*   __restrict__ src,
    const int*   __restrict__ dst,
    float*       __restrict__ gjT,     // (N,16) accumulator
    int edges)
{
    int e = blockIdx.x * blockDim.x + threadIdx.x;
    if (e >= edges) return;

    const int   s  = src[e];
    const int   d  = dst[e];
    const float ww = w[e];

    const f4* op4 = (const f4*)(o_preT + s * BATCH);
    const f4* et4 = (const f4*)(Et + d * BATCH);
    f4 oj[4], en[4];
#pragma unroll
    for (int k = 0; k < 4; ++k) { oj[k] = op4[k]; en[k] = et4[k]; }

    float* g = gjT + d * BATCH;
#pragma unroll
    for (int k = 0; k < 4; ++k) {
#pragma unroll
        for (int j = 0; j < 4; ++j) {
            float o = oj[k][j];
            float m = o * ww;
            float c = (o >= en[k][j]) ? m : -m;   // oj * w * sgn, bit-exact
            unsafeAtomicAdd(g + (4 * k + j), c);
        }
    }
}

// ---------------------------------------------------------------------------
// Zero a float4 array (gj accumulator must be re-zeroed every call: harness
// does not re-poison buffers between graph replays).
// ---------------------------------------------------------------------------
__global__ __launch_bounds__(256) void zero_f4(f4* __restrict__ p, int n4) {
    int t = blockIdx.x * blockDim.x + threadIdx.x;
    if (t < n4) {
        f4 z = {0.0f, 0.0f, 0.0f, 0.0f};
        p[t] = z;
    }
}

// ---------------------------------------------------------------------------
// Transpose (B=16, N) -> (N, 16) for E and o_pre in one pass.
// One thread per column n: 16 coalesced b32 reads, then a contiguous 64B
// write as 4x b128. 12.8MB total, L2-resident, negligible vs edge phase.
// ---------------------------------------------------------------------------
__global__ __launch_bounds__(256) void transpose16x2(
    const float* __restrict__ E, const float* __restrict__ o_pre,
    float* __restrict__ Et, float* __restrict__ o_preT, int N)
{
    int n = blockIdx.x * blockDim.x + threadIdx.x;
    if (n >= N) return;
    float e[BATCH], o[BATCH];
#pragma unroll
    for (int b = 0; b < BATCH; ++b) {
        e[b] = E[b * N + n];
        o[b] = o_pre[b * N + n];
    }
    f4* ed = (f4*)(Et + n * BATCH);
    f4* od = (f4*)(o_preT + n * BATCH);
#pragma unroll
    for (int k = 0; k < 4; ++k) {
        f4 ev = {e[4*k], e[4*k+1], e[4*k+2], e[4*k+3]};
        f4 ov = {o[4*k], o[4*k+1], o[4*k+2], o[4*k+3]};
        ed[k] = ev;
        od[k] = ov;
    }
}

// ---------------------------------------------------------------------------
// FAST PATH epilogue: float4 over (B,N) flat index; gj gathered from the
// transposed accumulator (4 scalar loads, stride 64B).
// ---------------------------------------------------------------------------
__global__ __launch_bounds__(256) void finalize_t(
    const f4* __restrict__ E4,
    const f4* __restrict__ chem4,
    const float* __restrict__ threshold,
    const float* __restrict__ decay,
    const float* __restrict__ gjT,     // (N,16)
    f4* __restrict__ out,              // [new_o | new_e] as f4
    int BN4, int N)
{
    int t = blockIdx.x * blockDim.x + threadIdx.x;
    if (t >= BN4) return;

    int i = t * 4;
    int n = i % N;
    int b = i / N;

    f4 e   = E4[t];
    f4 ch  = chem4[t];
    f4 thr = *(const f4*)(threshold + n);
    f4 dec = *(const f4*)(decay + n);
    f4 gj;
#pragma unroll
    for (int k = 0; k < 4; ++k) gj[k] = gjT[(n + k) * BATCH + b];

    f4 no, ne;
#pragma unroll
    for (int k = 0; k < 4; ++k) {
        float S = (e[k] + ch[k]) + gj[k];          // match reference add order
        S = fminf(fmaxf(S, CLAMP_MIN_F), CLAMP_MAX_F);
        bool  gt   = S > thr[k];
        bool  mask = (!gt) && (fabsf(S - e[k]) <= EPS_F);
        float no_k = fmaxf(S - thr[k], 0.0f);
        no[k] = no_k;
        ne[k] = gt ? no_k : (mask ? (e[k] - dec[k]) : S);
    }
    out[t]       = no;
    out[BN4 + t] = ne;
}

// ---------------------------------------------------------------------------
// FALLBACK PATH (ws too small): accumulator aliased into the new_e half of
// d_out (safe there: finalize reads its own slot before writing it).
// ---------------------------------------------------------------------------
__global__ __launch_bounds__(256) void edge_scatter(
    const float* __restrict__ o_pre, const float* __restrict__ E,
    const float* __restrict__ w, const int* __restrict__ src,
    const int* __restrict__ dst, float* __restrict__ gj,
    int edges, int N)
{
    int e = blockIdx.x * blockDim.x + threadIdx.x;
    if (e >= edges) return;
    const int   s  = src[e];
    const int   d  = dst[e];
    const float ww = w[e];
    float oj[BATCH], en[BATCH];
#pragma unroll
    for (int b = 0; b < BATCH; ++b) {
        oj[b] = o_pre[b * N + s];
        en[b] = E[b * N + d];
    }
#pragma unroll
    for (int b = 0; b < BATCH; ++b) {
        float m = oj[b] * ww;
        float c = (oj[b] >= en[b]) ? m : -m;
        unsafeAtomicAdd(gj + b * N + d, c);
    }
}

__global__ __launch_bounds__(256) void finalize(
    const f4* __restrict__ E4, const f4* __restrict__ chem4,
    const float* __restrict__ threshold, const float* __restrict__ decay,
    f4* __restrict__ out, int BN4, int N)
{
    int t = blockIdx.x * blockDim.x + threadIdx.x;
    if (t >= BN4) return;
    int i = t * 4;
    int n = i % N;
    f4 e   = E4[t];
    f4 ch  = chem4[t];
    f4 gj  = out[BN4 + t];
    f4 thr = *(const f4*)(threshold + n);
    f4 dec = *(const f4*)(decay + n);
    f4 no, ne;
#pragma unroll
    for (int k = 0; k < 4; ++k) {
        float S = (e[k] + ch[k]) + gj[k];
        S = fminf(fmaxf(S, CLAMP_MIN_F), CLAMP_MAX_F);
        bool  gt   = S > thr[k];
        bool  mask = (!gt) && (fabsf(S - e[k]) <= EPS_F);
        float no_k = fmaxf(S - thr[k], 0.0f);
        no[k] = no_k;
        ne[k] = gt ? no_k : (mask ? (e[k] - dec[k]) : S);
    }
    out[t]       = no;
    out[BN4 + t] = ne;
}

// ---------------------------------------------------------------------------
// Launch. Inputs in setup_inputs() order:
//   0 chem (B,N)  1 E (B,N)  2 o_pre (B,N)  3 w (EDGES)
//   4 threshold (N)  5 decay (N)  6 src (EDGES) i32  7 dst (EDGES) i32
// d_out = [new_o | new_e] flat (2*B*N f32).
// ---------------------------------------------------------------------------
extern "C" void kernel_launch(void* const* d_in, const int* in_sizes, int n_in,
                              void* d_out, int out_size, void* d_ws, size_t ws_size,
                              hipStream_t stream) {
    const float* chem      = (const float*)d_in[0];
    const float* E         = (const float*)d_in[1];
    const float* o_pre     = (const float*)d_in[2];
    const float* w         = (const float*)d_in[3];
    const float* threshold = (const float*)d_in[4];
    const float* decay     = (const float*)d_in[5];
    const int*   src       = (const int*)d_in[6];
    const int*   dst       = (const int*)d_in[7];

    const int BN    = in_sizes[0];   // B*N = 800000
    const int edges = in_sizes[3];   // 1600000
    const int N     = in_sizes[4];   // 50000
    const int BN4   = BN / 4;

    float* out = (float*)d_out;

    const int eb = (edges + 255) / 256;
    const int fb = (BN4 + 255) / 256;
    const int nb = (N + 255) / 256;

    const size_t need = 3ull * (size_t)BN * sizeof(float);
    if (ws_size >= need) {
        // --- fast path: transposed (N,16) operands + accumulator in ws ---
        float* o_preT = (float*)d_ws;
        float* Et     = o_preT + BN;
        float* gjT    = Et + BN;

        transpose16x2<<<nb, 256, 0, stream>>>(E, o_pre, Et, o_preT, N);
        zero_f4<<<fb, 256, 0, stream>>>((f4*)gjT, BN4);
        edge_scatter_t<<<eb, 256, 0, stream>>>(o_preT, Et, w, src, dst, gjT,
                                               edges);
        finalize_t<<<fb, 256, 0, stream>>>((const f4*)E, (const f4*)chem,
                                           threshold, decay, gjT, (f4*)out,
                                           BN4, N);
    } else {
        // --- fallback: accumulate in the new_e half of d_out ---
        float* gj = out + BN;
        zero_f4<<<fb, 256, 0, stream>>>((f4*)gj, BN4);
        edge_scatter<<<eb, 256, 0, stream>>>(o_pre, E, w, src, dst, gj,
                                             edges, N);
        finalize<<<fb, 256, 0, stream>>>((const f4*)E, (const f4*)chem,
                                         threshold, decay, (f4*)out, BN4, N);
    }
}